// clDiceLoss_25872882991527
// MI455X (gfx1250) — compile-verified
//
#include <hip/hip_runtime.h>
#include <math.h>
#include <stdint.h>

// Problem geometry (fixed by reference): B=16, C=1, H=W=512, 10 iterations.
#define IMG_H 512
#define IMG_W 512
#define BATCH 16
#define NPLANES 32                         // 2 chains (pred, target) x 16 batch
#define NELEM (BATCH * IMG_H * IMG_W)      // 4,194,304 per chain
#define PLANE_HW (IMG_H * IMG_W)
#define N_ITERS 10
#define SMOOTH_EPS 1e-6f

#define TILE 32
#define HALO 2
#define SIMG_W (TILE + 2 * HALO)           // 36
#define SIMG_P (SIMG_W + 1)                // 37 (pitch, avoid bank conflicts)
#define SERO_W (TILE + 2)                  // 34
#define SERO_P (SERO_W + 1)                // 35

#define RED_BLOCKS 1024
#define RED_THREADS 256

// gfx1250 async LDS path (ASYNCcnt-tracked, L2<->LDS without VGPR staging).
// Guarded so a missing builtin falls back to the proven synchronous path.
#if defined(__has_builtin)
#if __has_builtin(__builtin_amdgcn_global_load_async_to_lds_b32) && \
    __has_builtin(__builtin_amdgcn_global_store_async_from_lds_b32) && \
    __has_builtin(__builtin_amdgcn_s_wait_asynccnt)
#define USE_ASYNC_LDS 1
#endif
#endif

typedef __attribute__((ext_vector_type(2))) float v2f;
typedef __attribute__((ext_vector_type(8))) float v8f;

typedef __attribute__((address_space(3))) int as3_int;

__device__ __forceinline__ float dev_sigmoid(float x) {
    return 1.0f / (1.0f + __expf(-x));
}

// ---------------------------------------------------------------------------
// Init: img chain-p = sigmoid(pred), chain-t = target; skel = 0.
// ---------------------------------------------------------------------------
__global__ void cl_init(const float* __restrict__ pred,
                        const float* __restrict__ target,
                        float* __restrict__ img,
                        float* __restrict__ skel) {
    int stride = gridDim.x * blockDim.x;
    for (int i = blockIdx.x * blockDim.x + threadIdx.x; i < 2 * NELEM; i += stride) {
        float v = (i < NELEM) ? dev_sigmoid(pred[i]) : target[i - NELEM];
        img[i]  = v;
        skel[i] = 0.0f;
    }
}

// ---------------------------------------------------------------------------
// One fused soft-skeleton iteration over one 32x32 tile of one plane:
//   eroded = min(min3x1, min1x3)   (+inf padding == skip OOB)
//   opened = max3x3(eroded)        (-inf padding == skip OOB)
//   skel  += relu(img - opened)
//   imgOut = eroded
// ---------------------------------------------------------------------------
__global__ __launch_bounds__(256) void cl_iter(const float* __restrict__ imgIn,
                                               float* __restrict__ imgOut,
                                               float* __restrict__ skel) {
    __shared__ float simg[SIMG_W * SIMG_P];
    __shared__ float sero[SERO_W * SERO_P];

    const float INF = __builtin_inff();
    const int pz = blockIdx.z;
    const int ox = blockIdx.x * TILE;
    const int oy = blockIdx.y * TILE;
    const float* ip = imgIn + (size_t)pz * PLANE_HW;
    const int tid = threadIdx.y * 32 + threadIdx.x;

    // CDNA5 speculative prefetch of the tile's first cacheline (global_prefetch_b8)
    {
        int py = oy - HALO; if (py < 0) py = 0;
        int px = ox - HALO; if (px < 0) px = 0;
        __builtin_prefetch(ip + (size_t)py * IMG_W + px, 0, 3);
    }

    // Load 36x36 img tile; +inf outside the image (identity for min).
    for (int k = tid; k < SIMG_W * SIMG_W; k += 256) {
        int r = k / SIMG_W, c = k - r * SIMG_W;
        int gy = oy - HALO + r, gx = ox - HALO + c;
        if (gy >= 0 && gy < IMG_H && gx >= 0 && gx < IMG_W) {
#ifdef USE_ASYNC_LDS
            __builtin_amdgcn_global_load_async_to_lds_b32(
                (int*)(ip + gy * IMG_W + gx),
                (as3_int*)&simg[r * SIMG_P + c],
                /*offset=*/0, /*cpol=*/0);
#else
            simg[r * SIMG_P + c] = ip[gy * IMG_W + gx];
#endif
        } else {
            simg[r * SIMG_P + c] = INF;
        }
    }
#ifdef USE_ASYNC_LDS
    __builtin_amdgcn_s_wait_asynccnt(0);
#endif
    __syncthreads();

    // Eroded 34x34 region: local (er,ec) <-> global (oy-1+er, ox-1+ec).
    // Positions outside the image get -inf (identity for the following max).
    for (int k = tid; k < SERO_W * SERO_W; k += 256) {
        int er = k / SERO_W, ec = k - er * SERO_W;
        int gy = oy - 1 + er, gx = ox - 1 + ec;
        float v;
        if (gy >= 0 && gy < IMG_H && gx >= 0 && gx < IMG_W) {
            const float* s = &simg[er * SIMG_P + ec];
            float c0 = s[1];                 // (gy-1, gx)
            float c1 = s[SIMG_P + 1];        // (gy  , gx)
            float c2 = s[2 * SIMG_P + 1];    // (gy+1, gx)
            float vmin = fminf(fminf(c0, c1), c2);
            float h0 = s[SIMG_P + 0];        // (gy, gx-1)
            float h2 = s[SIMG_P + 2];        // (gy, gx+1)
            float hmin = fminf(fminf(h0, c1), h2);
            v = fminf(vmin, hmin);
        } else {
            v = -INF;
        }
        sero[er * SERO_P + ec] = v;
    }
    __syncthreads();

    // 32x32 outputs, each thread handles 4 rows.
    const size_t base = (size_t)pz * PLANE_HW;
    #pragma unroll
    for (int q = 0; q < 4; q++) {
        int r = threadIdx.y + 8 * q;
        int c = threadIdx.x;
        float m = -INF;
        #pragma unroll
        for (int dr = 0; dr < 3; dr++) {
            const float* s = &sero[(r + dr) * SERO_P + c];
            m = fmaxf(m, fmaxf(fmaxf(s[0], s[1]), s[2]));
        }
        float center = simg[(r + HALO) * SIMG_P + (c + HALO)];
        size_t idx = base + (size_t)(oy + r) * IMG_W + (ox + c);
        float d = center - m;
        skel[idx] += (d > 0.0f ? d : 0.0f);
        // eroded center goes straight from LDS to L2 (async store, ASYNCcnt;
        // s_endpgm's implicit wait-idle drains it). Fallback: VGPR store.
#ifdef USE_ASYNC_LDS
        __builtin_amdgcn_global_store_async_from_lds_b32(
            (int*)(imgOut + idx),
            (as3_int*)&sero[(r + 1) * SERO_P + (c + 1)],
            /*offset=*/0, /*cpol=*/0);
#else
        imgOut[idx] = sero[(r + 1) * SERO_P + (c + 1)];
#endif
    }
}

// ---------------------------------------------------------------------------
// Reduction via V_WMMA_F32_16X16X4_F32 (full-fp32 matrix op, f32 accumulate).
// A = ones(16x4)  =>  D[m][n] = sum_k B[k][n]; summing c[0] across all 32
// lanes gives row0+row8 column sums = 2 * (sum of the 64 values fed as B).
// Four accumulators: S0=Σ p·skel_gt, S1=Σ skel_gt, S2=Σ t·skel_pred, S3=Σ skel_pred.
// ---------------------------------------------------------------------------
__global__ __launch_bounds__(RED_THREADS) void cl_reduce(const float* __restrict__ pred,
                                                         const float* __restrict__ target,
                                                         const float* __restrict__ skel,
                                                         float* __restrict__ partials) {
    const v2f ones = {1.0f, 1.0f};
    v8f acc0 = {}, acc1 = {}, acc2 = {}, acc3 = {};

    const int lane   = threadIdx.x & 31;
    const int gwave  = (blockIdx.x * blockDim.x + threadIdx.x) >> 5;
    const int nwaves = (gridDim.x * blockDim.x) >> 5;

    // NELEM = 65536 * 64; trip count is identical for every wave -> EXEC all-1s.
    for (size_t chunk = (size_t)gwave * 64; chunk < (size_t)NELEM; chunk += (size_t)nwaves * 64) {
        size_t i0 = chunk + lane;
        size_t i1 = chunk + lane + 32;
        float p0  = dev_sigmoid(pred[i0]);
        float p1  = dev_sigmoid(pred[i1]);
        float t0  = target[i0], t1 = target[i1];
        float sp0 = skel[i0],   sp1 = skel[i1];                 // skel_pred chain
        float sg0 = skel[NELEM + i0], sg1 = skel[NELEM + i1];   // skel_gt chain

        v2f b0 = {p0 * sg0, p1 * sg1};
        v2f b1 = {sg0, sg1};
        v2f b2 = {t0 * sp0, t1 * sp1};
        v2f b3 = {sp0, sp1};

        acc0 = __builtin_amdgcn_wmma_f32_16x16x4_f32(false, ones, false, b0, (short)0, acc0, false, false);
        acc1 = __builtin_amdgcn_wmma_f32_16x16x4_f32(false, ones, false, b1, (short)0, acc1, false, false);
        acc2 = __builtin_amdgcn_wmma_f32_16x16x4_f32(false, ones, false, b2, (short)0, acc2, false, false);
        acc3 = __builtin_amdgcn_wmma_f32_16x16x4_f32(false, ones, false, b3, (short)0, acc3, false, false);
    }

    // Per-wave: butterfly-reduce c[0] across 32 lanes, halve (row0+row8 both total).
    float vals[4] = {acc0[0], acc1[0], acc2[0], acc3[0]};
    #pragma unroll
    for (int j = 0; j < 4; j++) {
        float v = vals[j];
        #pragma unroll
        for (int off = 16; off > 0; off >>= 1)
            v += __shfl_xor(v, off, 32);
        vals[j] = v * 0.5f;
    }

    __shared__ float red[8][4];
    int wib = threadIdx.x >> 5;
    if (lane == 0) {
        red[wib][0] = vals[0];
        red[wib][1] = vals[1];
        red[wib][2] = vals[2];
        red[wib][3] = vals[3];
    }
    __syncthreads();
    if (threadIdx.x < 4) {
        float s = 0.0f;
        #pragma unroll
        for (int w = 0; w < 8; w++) s += red[w][threadIdx.x];
        partials[blockIdx.x * 4 + threadIdx.x] = s;  // deterministic (no atomics)
    }
}

// ---------------------------------------------------------------------------
// Final scalar combine (deterministic serial sums; ~4K adds).
// ---------------------------------------------------------------------------
__global__ void cl_finalize(const float* __restrict__ partials, int nblocks,
                            float* __restrict__ out) {
    if (threadIdx.x == 0 && blockIdx.x == 0) {
        float s0 = 0.f, s1 = 0.f, s2 = 0.f, s3 = 0.f;
        for (int b = 0; b < nblocks; b++) {
            s0 += partials[b * 4 + 0];
            s1 += partials[b * 4 + 1];
            s2 += partials[b * 4 + 2];
            s3 += partials[b * 4 + 3];
        }
        float tprec  = s0 / (s1 + SMOOTH_EPS);
        float tsens  = s2 / (s3 + SMOOTH_EPS);
        float cldice = 2.0f * tprec * tsens / (tprec + tsens + SMOOTH_EPS);
        out[0] = 1.0f - cldice;
    }
}

// ---------------------------------------------------------------------------
// Workspace layout (floats):
//   [0,        2N)  imgA (ping)    chain-p then chain-t
//   [2N,       4N)  imgB (pong)
//   [4N,       6N)  skel           skel_pred then skel_gt
//   [6N, 6N+4*RB)   reduction partials
// Total ~100.7 MB — fits comfortably in MI455X's 192 MB L2.
// ---------------------------------------------------------------------------
extern "C" void kernel_launch(void* const* d_in, const int* in_sizes, int n_in,
                              void* d_out, int out_size, void* d_ws, size_t ws_size,
                              hipStream_t stream) {
    const float* pred   = (const float*)d_in[0];
    const float* target = (const float*)d_in[1];
    float* ws   = (float*)d_ws;
    float* imgA = ws;
    float* imgB = ws + 2 * (size_t)NELEM;
    float* skel = ws + 4 * (size_t)NELEM;
    float* partials = ws + 6 * (size_t)NELEM;

    cl_init<<<4096, 256, 0, stream>>>(pred, target, imgA, skel);

    dim3 grid(IMG_W / TILE, IMG_H / TILE, NPLANES);
    dim3 block(32, 8);
    float* cur = imgA;
    float* nxt = imgB;
    for (int it = 0; it < N_ITERS; it++) {
        cl_iter<<<grid, block, 0, stream>>>(cur, nxt, skel);
        float* t = cur; cur = nxt; nxt = t;
    }

    cl_reduce<<<RED_BLOCKS, RED_THREADS, 0, stream>>>(pred, target, skel, partials);
    cl_finalize<<<1, 32, 0, stream>>>(partials, RED_BLOCKS, (float*)d_out);
}